// BRepGAT_69234872812152
// MI455X (gfx1250) — compile-verified
//
#include <hip/hip_runtime.h>
#include <hip/hip_bf16.h>

typedef __attribute__((ext_vector_type(2))) float v2f;
typedef __attribute__((ext_vector_type(8))) float v8f;

#define TPB 256

// ---------------------------------------------------------------------------
// WMMA f32 GEMM: H[M x outc] = X[M x K] @ W[K x outc]
// One wave owns a 16-row strip and CT 16-col tiles: the A fragment loaded per
// K-step is reused by CT v_wmma_f32_16x16x4_f32 issues (independent
// accumulators -> co-executable back-to-back WMMAs).
// ---------------------------------------------------------------------------
template <int CT>
__global__ __launch_bounds__(32)
void gemm_wmma_f32_kernel(const float* __restrict__ X, const float* __restrict__ W,
                          float* __restrict__ H, int K, int outc) {
    int rt = blockIdx.x;
    int lane = threadIdx.x & 31;
    int half = lane >> 4;      // 0: K pair {0,1} / M rows 0..7 ; 1: K pair {2,3} / M rows 8..15
    int l16  = lane & 15;
    int m = rt * 16 + l16;     // A-matrix row for this lane

    v8f acc[CT] = {};
    for (int k0 = 0; k0 < K; k0 += 4) {
        int ka = k0 + half * 2;
        v2f a;
        a.x = X[(size_t)m * K + ka];
        a.y = X[(size_t)m * K + ka + 1];
#pragma unroll
        for (int ct = 0; ct < CT; ++ct) {
            int n = ct * 16 + l16;
            v2f b;
            b.x = W[(size_t)ka * outc + n];
            b.y = W[(size_t)(ka + 1) * outc + n];
            acc[ct] = __builtin_amdgcn_wmma_f32_16x16x4_f32(
                /*neg_a=*/false, a, /*neg_b=*/false, b,
                /*c_mod=*/(short)0, acc[ct], /*reuse_a=*/false, /*reuse_b=*/false);
        }
    }
#pragma unroll
    for (int ct = 0; ct < CT; ++ct) {
        int n = ct * 16 + l16;
#pragma unroll
        for (int r = 0; r < 8; ++r) {
            int mm = rt * 16 + r + half * 8;
            H[(size_t)mm * outc + n] = acc[ct][r];
        }
    }
}

// ---------------------------------------------------------------------------
// Utility fill
// ---------------------------------------------------------------------------
__global__ void fill_kernel(float* __restrict__ p, float v, int n) {
    int i = blockIdx.x * blockDim.x + threadIdx.x;
    if (i < n) p[i] = v;
}

// ---------------------------------------------------------------------------
// Self-loop attr: mean of incoming edge_attr per node
// ---------------------------------------------------------------------------
__global__ void loop_scatter_kernel(const int* __restrict__ dst, const float* __restrict__ eattr,
                                    float* __restrict__ cnt, float* __restrict__ lacc, int E_) {
    int e = blockIdx.x * blockDim.x + threadIdx.x;
    if (e >= E_) return;
    int d = dst[e];
    atomicAdd(&cnt[d], 1.0f);
#pragma unroll
    for (int k = 0; k < 16; ++k)
        atomicAdd(&lacc[(size_t)d * 16 + k], eattr[(size_t)e * 16 + k]);
}

__global__ void loop_div_kernel(float* __restrict__ lacc, const float* __restrict__ cnt, int N_) {
    int i = blockIdx.x * blockDim.x + threadIdx.x;
    if (i >= N_) return;
    float c = cnt[i];
    float inv = 1.0f / (c > 1.0f ? c : 1.0f);
#pragma unroll
    for (int k = 0; k < 16; ++k) lacc[(size_t)i * 16 + k] *= inv;
}

// ---------------------------------------------------------------------------
// ve[k][h] = sum_c We[k, h*16+c] * ae[h*16+c]   (folds edge projection)
// ---------------------------------------------------------------------------
__global__ void ve_kernel(const float* __restrict__ We, const float* __restrict__ ae,
                          float* __restrict__ ve, int Hh) {
    int t = threadIdx.x;
    if (t >= 16 * Hh) return;
    int k = t / Hh, h = t % Hh;
    int outc = Hh * 16;
    float s = 0.f;
#pragma unroll
    for (int c = 0; c < 16; ++c) s += We[k * outc + h * 16 + c] * ae[h * 16 + c];
    ve[k * Hh + h] = s;
}

// ---------------------------------------------------------------------------
// Per-node attention coefficients: alpha_s/alpha_d = (h * a).sum(-1)
// ---------------------------------------------------------------------------
__global__ void node_alpha_kernel(const float* __restrict__ hbuf,
                                  const float* __restrict__ as_, const float* __restrict__ ad_,
                                  float* __restrict__ als, float* __restrict__ ald,
                                  int N_, int Hh) {
    int n = blockIdx.x * blockDim.x + threadIdx.x;
    if (n >= N_) return;
    int outc = Hh * 16;
    for (int h = 0; h < Hh; ++h) {
        float ss = 0.f, dd = 0.f;
#pragma unroll
        for (int c = 0; c < 16; ++c) {
            float hv = hbuf[(size_t)n * outc + h * 16 + c];
            ss += hv * as_[h * 16 + c];
            dd += hv * ad_[h * 16 + c];
        }
        als[n * Hh + h] = ss;
        ald[n * Hh + h] = dd;
    }
}

// ---------------------------------------------------------------------------
// Float atomic max via signed/unsigned bit trick (requires -inf init)
// ---------------------------------------------------------------------------
__device__ inline void atomicMaxF(float* addr, float v) {
    if (v >= 0.0f) atomicMax((int*)addr, __float_as_int(v));
    else           atomicMin((unsigned int*)addr, __float_as_uint(v));
}

// ---------------------------------------------------------------------------
// Edge alpha = leaky_relu(as[src] + ad[dst] + eattr.ve); segment max over dst
// ---------------------------------------------------------------------------
__global__ void edge_alpha_kernel(const int* __restrict__ src, const int* __restrict__ dst,
                                  const float* __restrict__ eattr, const float* __restrict__ lacc,
                                  const float* __restrict__ ve,
                                  const float* __restrict__ als, const float* __restrict__ ald,
                                  float* __restrict__ ealpha, float* __restrict__ amax,
                                  int E_, int N_, int Hh) {
    int e = blockIdx.x * blockDim.x + threadIdx.x;
    int E2 = E_ + N_;
    if (e >= E2) return;
    int s, d;
    const float* ea;
    if (e < E_) { s = src[e]; d = dst[e]; ea = eattr + (size_t)e * 16; }
    else        { s = d = e - E_;          ea = lacc  + (size_t)(e - E_) * 16; }
    float er[16];
#pragma unroll
    for (int k = 0; k < 16; ++k) er[k] = ea[k];
    for (int h = 0; h < Hh; ++h) {
        float aE = 0.f;
#pragma unroll
        for (int k = 0; k < 16; ++k) aE += er[k] * ve[k * Hh + h];
        float al = als[s * Hh + h] + ald[d * Hh + h] + aE;
        al = al > 0.f ? al : 0.2f * al;           // leaky_relu(0.2)
        ealpha[(size_t)e * Hh + h] = al;
        atomicMaxF(&amax[d * Hh + h], al);
    }
}

// ---------------------------------------------------------------------------
// Fused softmax + aggregation (single edge pass, thread per edge*head):
//   ex = exp(alpha - amax[dst]); den[dst,h] += ex; obuf[dst,h,:] += ex*h[src,h,:]
// Normalization by den happens in node_finish (algebraically identical).
// ---------------------------------------------------------------------------
__global__ void edge_softmax_aggr_kernel(const int* __restrict__ src, const int* __restrict__ dst,
                                         const float* __restrict__ ealpha,
                                         const float* __restrict__ amax, float* __restrict__ den,
                                         const float* __restrict__ hbuf, float* __restrict__ obuf,
                                         int E_, int N_, int Hh) {
    int idx = blockIdx.x * blockDim.x + threadIdx.x;
    int total = (E_ + N_) * Hh;
    if (idx >= total) return;
    int e = idx / Hh;
    int h = idx - e * Hh;
    int s, d;
    if (e < E_) { s = src[e]; d = dst[e]; }
    else        { s = d = e - E_; }
    int outc = Hh * 16;
    float ex = __expf(ealpha[idx] - amax[d * Hh + h]);
    atomicAdd(&den[d * Hh + h], ex);
    const float* hs = hbuf + (size_t)s * outc + h * 16;
    float*       ob = obuf + (size_t)d * outc + h * 16;
#pragma unroll
    for (int c = 0; c < 16; ++c) atomicAdd(&ob[c], hs[c] * ex);
}

// ---------------------------------------------------------------------------
// out = obuf/(den+eps) + b, optional ReLU (in place)
// ---------------------------------------------------------------------------
__global__ void node_finish_kernel(float* __restrict__ obuf, const float* __restrict__ den,
                                   const float* __restrict__ b,
                                   int N_, int outc, int Hh, int do_relu) {
    int i = blockIdx.x * blockDim.x + threadIdx.x;
    if (i >= N_ * outc) return;
    int col = i % outc;
    int n   = i / outc;
    int h   = col >> 4;
    float v = obuf[i] / (den[n * Hh + h] + 1e-16f) + b[col];
    if (do_relu) v = v > 0.f ? v : 0.f;
    obuf[i] = v;
}

// ---------------------------------------------------------------------------
// Final MLP: relu(x@Wm1+bm1)@Wm2+bm2   (x stride 16)
// ---------------------------------------------------------------------------
__global__ void mlp_kernel(const float* __restrict__ x,
                           const float* __restrict__ Wm1, const float* __restrict__ bm1,
                           const float* __restrict__ Wm2, const float* __restrict__ bm2,
                           float* __restrict__ out, int N_) {
    int n = blockIdx.x * blockDim.x + threadIdx.x;
    if (n >= N_) return;
    float xr[16], hr[16];
#pragma unroll
    for (int k = 0; k < 16; ++k) xr[k] = x[(size_t)n * 16 + k];
#pragma unroll
    for (int j = 0; j < 16; ++j) {
        float s = bm1[j];
#pragma unroll
        for (int k = 0; k < 16; ++k) s += xr[k] * Wm1[k * 16 + j];
        hr[j] = s > 0.f ? s : 0.f;
    }
    for (int j = 0; j < 25; ++j) {
        float s = bm2[j];
#pragma unroll
        for (int k = 0; k < 16; ++k) s += hr[k] * Wm2[k * 25 + j];
        out[(size_t)n * 25 + j] = s;
    }
}

// ---------------------------------------------------------------------------
extern "C" void kernel_launch(void* const* d_in, const int* in_sizes, int n_in,
                              void* d_out, int out_size, void* d_ws, size_t ws_size,
                              hipStream_t stream) {
    const float* x0    = (const float*)d_in[0];
    const int*   eidx  = (const int*)d_in[1];
    const float* eattr = (const float*)d_in[2];

    const int N_ = in_sizes[0] / 64;          // 50000
    const int E_ = in_sizes[1] / 2;           // 800000
    const int E2 = E_ + N_;
    const int* src = eidx;
    const int* dst = eidx + E_;

    const float* Wm1 = (const float*)d_in[33];
    const float* bm1 = (const float*)d_in[34];
    const float* Wm2 = (const float*)d_in[35];
    const float* bm2 = (const float*)d_in[36];
    float* out = (float*)d_out;

    // workspace layout (floats)
    float* ws = (float*)d_ws;
    size_t o = 0;
    float* xA     = ws + o; o += (size_t)N_ * 64;
    float* xB     = ws + o; o += (size_t)N_ * 64;
    float* hbuf   = ws + o; o += (size_t)N_ * 64;
    float* lacc   = ws + o; o += (size_t)N_ * 16;
    float* cnt    = ws + o; o += (size_t)N_;
    float* als    = ws + o; o += (size_t)N_ * 4;
    float* ald    = ws + o; o += (size_t)N_ * 4;
    float* amax   = ws + o; o += (size_t)N_ * 4;
    float* den    = ws + o; o += (size_t)N_ * 4;
    float* ealpha = ws + o; o += (size_t)E2 * 4;
    float* ve     = ws + o; o += 64;
    (void)ws_size; (void)n_in; (void)out_size;

    const float NEG_INF = -__builtin_huge_valf();
    #define NB(n) ((unsigned)(((n) + TPB - 1) / TPB))

    // self-loop mean edge attrs
    fill_kernel<<<NB(N_), TPB, 0, stream>>>(cnt, 0.f, N_);
    fill_kernel<<<NB(N_ * 16), TPB, 0, stream>>>(lacc, 0.f, N_ * 16);
    loop_scatter_kernel<<<NB(E_), TPB, 0, stream>>>(dst, eattr, cnt, lacc, E_);
    loop_div_kernel<<<NB(N_), TPB, 0, stream>>>(lacc, cnt, N_);

    const int Hs[5] = {4, 4, 4, 4, 1};
    const float* xcur = x0;
    float* obufs[5] = {xA, xB, xA, xB, xA};
    const int rowTiles = N_ / 16;             // 50000 % 16 == 0

    for (int l = 0; l < 5; ++l) {
        int base = 3 + l * 6;
        const float* W   = (const float*)d_in[base + 0];
        const float* We  = (const float*)d_in[base + 1];
        const float* a_s = (const float*)d_in[base + 2];
        const float* a_d = (const float*)d_in[base + 3];
        const float* a_e = (const float*)d_in[base + 4];
        const float* b   = (const float*)d_in[base + 5];
        int Hh = Hs[l];
        int outc = Hh * 16;
        float* obuf = obufs[l];

        // h = x @ W  (V_WMMA_F32_16X16X4_F32, A fragment reused across col tiles)
        if (Hh == 4)
            gemm_wmma_f32_kernel<4><<<(unsigned)rowTiles, 32, 0, stream>>>(xcur, W, hbuf, 64, outc);
        else
            gemm_wmma_f32_kernel<1><<<(unsigned)rowTiles, 32, 0, stream>>>(xcur, W, hbuf, 64, outc);

        ve_kernel<<<1, 16 * Hh, 0, stream>>>(We, a_e, ve, Hh);
        node_alpha_kernel<<<NB(N_), TPB, 0, stream>>>(hbuf, a_s, a_d, als, ald, N_, Hh);

        fill_kernel<<<NB(N_ * Hh), TPB, 0, stream>>>(amax, NEG_INF, N_ * Hh);
        fill_kernel<<<NB(N_ * Hh), TPB, 0, stream>>>(den, 0.f, N_ * Hh);
        fill_kernel<<<NB(N_ * outc), TPB, 0, stream>>>(obuf, 0.f, N_ * outc);

        edge_alpha_kernel<<<NB(E2), TPB, 0, stream>>>(src, dst, eattr, lacc, ve,
                                                      als, ald, ealpha, amax, E_, N_, Hh);
        edge_softmax_aggr_kernel<<<NB(E2 * Hh), TPB, 0, stream>>>(src, dst, ealpha, amax, den,
                                                                  hbuf, obuf, E_, N_, Hh);
        node_finish_kernel<<<NB(N_ * outc), TPB, 0, stream>>>(obuf, den, b, N_, outc, Hh,
                                                              l < 4 ? 1 : 0);
        xcur = obuf;
    }

    mlp_kernel<<<NB(N_), TPB, 0, stream>>>(xcur, Wm1, bm1, Wm2, bm2, out, N_);
    #undef NB
}